// AudioQuantizer_23132693856659
// MI455X (gfx1250) — compile-verified
//
#include <hip/hip_runtime.h>

typedef __attribute__((ext_vector_type(2))) float v2f;
typedef __attribute__((ext_vector_type(4))) float v4f;
typedef __attribute__((ext_vector_type(8))) float v8f;

#define D_DIM     512
#define N_CB      8192
#define B_TOK     16384
#define KSTEPS    (D_DIM / 4)     // 128 wmma K-steps of 4
#define RES_JG    4               // K-groups (of 4 steps) resident in VGPRs
#define RES_STEPS (RES_JG * 4)    // 16 resident K-steps -> 32 VGPRs
#define LDS_JG    (32 - RES_JG)   // 28 K-groups streamed from LDS
#define SLAB_F    (LDS_JG * 256)  // 7168 floats per wave slab (28 KB)
#define WAVES_PB  4               // 4 waves / 128 threads per block
#define LDS_BYTES (WAVES_PB * LDS_JG * 1024)   // 112 KB per block
#define PIPE      8               // B chunks in flight: 32 loads, 128 VGPRs

static __device__ __forceinline__ v2f lo2(v4f v) { return __builtin_shufflevector(v, v, 0, 1); }
static __device__ __forceinline__ v2f hi2(v4f v) { return __builtin_shufflevector(v, v, 2, 3); }

#define WMMA_F32(A, Bv, C) \
    __builtin_amdgcn_wmma_f32_16x16x4_f32(false, (A), false, (Bv), (short)0, (C), false, false)

// ---------------------------------------------------------------------------
// Kernel 0a: hc2[n] = 0.5 * sum_k codebook[n][k]^2   (one wave32 per row)
// argmax(x.c - hc2) == argmin ||x-c||^2 ; the x^2 term is row-constant.
// ---------------------------------------------------------------------------
__global__ void c2_kernel(const float* __restrict__ cb, float* __restrict__ hc2) {
    int row  = (int)((blockIdx.x * blockDim.x + threadIdx.x) >> 5);
    int lane = threadIdx.x & 31;
    if (row >= N_CB) return;
    const float* r = cb + (size_t)row * D_DIM;
    float s = 0.0f;
    #pragma unroll
    for (int k = 0; k < D_DIM / 32; ++k) {
        float v = r[lane + 32 * k];
        s += v * v;
    }
    #pragma unroll
    for (int off = 16; off >= 1; off >>= 1)
        s += __shfl_xor(s, off, 32);
    if (lane == 0) hc2[row] = 0.5f * s;
}

// ---------------------------------------------------------------------------
// Kernel 0b: pre-swizzle codebook into WMMA B-fragment order.
// Fragment (tile t, K-step j=4*jg+q, lane L) is the v2f
//   cb[(16t + (L&15))*512 + 4j + (L>>4)*2]
// stored at swb[t*8192 + jg*256 + L*8 + q*2] so each lane reads TWO K-steps
// per 128b load, 1 KB contiguous per wave per group (perfect coalescing).
// ---------------------------------------------------------------------------
__global__ void swizzle_cb_kernel(const float* __restrict__ cb, float* __restrict__ swb) {
    unsigned tid = blockIdx.x * 256u + threadIdx.x;   // 2,097,152 fragments
    int q  = tid & 3;
    int L  = (tid >> 2) & 31;
    int jg = (tid >> 7) & 31;
    int t  = (int)(tid >> 12);
    int n  = (t << 4) + (L & 15);
    int k  = (jg << 4) + (q << 2) + ((L >> 4) << 1);
    v2f v = *(const v2f*)(cb + (size_t)n * D_DIM + k);
    *(v2f*)(swb + 2 * (size_t)tid) = v;
}

// ---------------------------------------------------------------------------
// Kernel 1: fused f32-WMMA GEMM + running argmin, with an explicit
// modulo-scheduled software pipeline for the B stream:
//  - rotating PIPE=8 chunk buffer (chunk = 4x v4f = both 16-col tiles of one
//    K-group), load for chunk jg+8 issued while consuming chunk jg
//    -> 32 loads permanently in flight, ~64-wmma prefetch distance.
//  - pipeline flows across t2 iterations via immediate offsets (+16384 fl).
//  - sched_group_barrier {8 wmma, 2 ds-read, 4 vmem-read} per unit keeps the
//    scheduler from sinking prefetches back to their consumers.
// ---------------------------------------------------------------------------
__launch_bounds__(128, 1)
__global__ void vq_argmin_kernel(const float* __restrict__ x,
                                 const float* __restrict__ swb,
                                 const float* __restrict__ hc2,
                                 int* __restrict__ idx_out) {
    extern __shared__ float xlds[];
    const int lane  = threadIdx.x & 31;
    const int wave  = threadIdx.x >> 5;          // 0..3
    const int m0    = blockIdx.x * 64 + wave * 16;
    const int mlane = lane & 15;                 // M (A) / N (B,C) sub-lane
    const int khalf = lane >> 4;                 // 0: K={0,1}  1: K={2,3}

    const float* xr = x + (size_t)(m0 + mlane) * D_DIM + khalf * 2;

    // ---- resident A fragments: K-steps 0..15 (32 VGPRs) ----
    v2f a[RES_STEPS];
    #pragma unroll
    for (int j = 0; j < RES_STEPS; ++j)
        a[j] = *(const v2f*)(xr + 4 * j);

    // ---- stage K-steps 16..127 into LDS, lane-major / b128-friendly ----
    float* slab = xlds + wave * SLAB_F;
    #pragma unroll
    for (int j = RES_STEPS; j < KSTEPS; ++j) {
        v2f v = *(const v2f*)(xr + 4 * j);
        int g = (j >> 2) - RES_JG;
        *(v2f*)(slab + g * 256 + lane * 8 + (j & 3) * 2) = v;
    }
    __syncthreads();

    float best[8];
    int   bidx[8];
    #pragma unroll
    for (int r = 0; r < 8; ++r) { best[r] = -3.0e38f; bidx[r] = 0; }

    const float* slr = slab + lane * 8;

    // ---- software-pipeline prologue: chunks 0..7 of iteration 0 ----
    v4f B01[PIPE], B23[PIPE], C01[PIPE], C23[PIPE];
    {
        const float* bpP = swb + lane * 8;
        #pragma unroll
        for (int s = 0; s < PIPE; ++s) {
            B01[s] = *(const v4f*)(bpP + s * 256);
            B23[s] = *(const v4f*)(bpP + s * 256 + 4);
            C01[s] = *(const v4f*)(bpP + 8192 + s * 256);
            C23[s] = *(const v4f*)(bpP + 8192 + s * 256 + 4);
        }
    }

    for (int t2 = 0; t2 < N_CB / 32; ++t2) {     // 256 iterations, 32 cols each
        asm volatile("" ::: "memory");   // keep loop-invariant LDS/A reads in-loop
        const float* bp0 = swb + (size_t)t2 * 16384 + lane * 8;
        const int    n0  = (t2 << 5) + mlane;
        const int    n1  = n0 + 16;
        const float  hc0 = hc2[n0];
        const float  hc1 = hc2[n1];

        v8f p0 = {}, p1 = {}, q0 = {}, q1 = {};  // 4 independent wmma chains

        #pragma unroll
        for (int jg = 0; jg < 32; ++jg) {
            const int s = jg & (PIPE - 1);

            v2f a0, a1, a2, a3;
            if (jg < RES_JG) {
                a0 = a[4 * jg + 0]; a1 = a[4 * jg + 1];
                a2 = a[4 * jg + 2]; a3 = a[4 * jg + 3];
            } else {
                v4f a01 = *(const v4f*)(slr + (jg - RES_JG) * 256);
                v4f a23 = *(const v4f*)(slr + (jg - RES_JG) * 256 + 4);
                a0 = lo2(a01); a1 = hi2(a01);
                a2 = lo2(a23); a3 = hi2(a23);
            }

            p0 = WMMA_F32(a0, lo2(B01[s]), p0);
            p1 = WMMA_F32(a1, hi2(B01[s]), p1);
            q0 = WMMA_F32(a0, lo2(C01[s]), q0);
            q1 = WMMA_F32(a1, hi2(C01[s]), q1);
            p0 = WMMA_F32(a2, lo2(B23[s]), p0);
            p1 = WMMA_F32(a3, hi2(B23[s]), p1);
            q0 = WMMA_F32(a2, lo2(C23[s]), q0);
            q1 = WMMA_F32(a3, hi2(C23[s]), q1);

            // prefetch chunk jg+PIPE (spills into next t2 at +16384 floats)
            const int    pc = jg + PIPE;
            const float* pb = (pc < 32) ? (bp0 + pc * 256)
                                        : (bp0 + 16384 + (pc - 32) * 256);
            B01[s] = *(const v4f*)(pb);
            B23[s] = *(const v4f*)(pb + 4);
            C01[s] = *(const v4f*)(pb + 8192);
            C23[s] = *(const v4f*)(pb + 8192 + 4);

            // pin the unit shape: {8 wmma, 2 ds-read, 4 vmem-read}
            __builtin_amdgcn_sched_group_barrier(0x008, 8, 0);
            __builtin_amdgcn_sched_group_barrier(0x100, 2, 0);
            __builtin_amdgcn_sched_group_barrier(0x020, 4, 0);
        }

        v8f s0 = p0 + p1;
        v8f s1 = q0 + q1;
        #pragma unroll
        for (int r = 0; r < 8; ++r) {            // n0 first: ties -> lowest n
            float sc0 = s0[r] - hc0;
            if (sc0 > best[r]) { best[r] = sc0; bidx[r] = n0; }
            float sc1 = s1[r] - hc1;
            if (sc1 > best[r]) { best[r] = sc1; bidx[r] = n1; }
        }
    }

    // ---- per-row argmax across the 16 lanes of each half-wave ----
    #pragma unroll
    for (int r = 0; r < 8; ++r) {
        float v = best[r];
        int   i = bidx[r];
        #pragma unroll
        for (int off = 1; off < 16; off <<= 1) {
            float ov = __shfl_xor(v, off, 32);
            int   oi = __shfl_xor(i, off, 32);
            if (ov > v || (ov == v && oi < i)) { v = ov; i = oi; }  // ties: lowest n
        }
        if (mlane == 0)
            idx_out[m0 + r + khalf * 8] = i;
    }
}

// ---------------------------------------------------------------------------
// Kernel 2: out[m][:] = embedding[idx[m]][:]   (float4 row gather)
// ---------------------------------------------------------------------------
__global__ void gather_kernel(const float* __restrict__ emb,
                              const int* __restrict__ idx,
                              float* __restrict__ out) {
    size_t tid = (size_t)blockIdx.x * blockDim.x + threadIdx.x;  // B_TOK*128 float4s
    int row = (int)(tid >> 7);
    int col = (int)(tid & 127);
    const float4* src = (const float4*)(emb + (size_t)idx[row] * D_DIM) + col;
    float4*       dst = (float4*)(out + (size_t)row * D_DIM) + col;
    *dst = *src;
}

// ---------------------------------------------------------------------------
extern "C" void kernel_launch(void* const* d_in, const int* in_sizes, int n_in,
                              void* d_out, int out_size, void* d_ws, size_t ws_size,
                              hipStream_t stream) {
    (void)in_sizes; (void)n_in; (void)out_size; (void)ws_size;
    const float* x   = (const float*)d_in[0];   // (B, D)
    const float* cb  = (const float*)d_in[1];   // (N, D)
    const float* emb = (const float*)d_in[2];   // (N, D)
    float* out = (float*)d_out;                 // (B, D)

    float* swb = (float*)d_ws;                                     // 16 MB swizzled B
    float* hc2 = (float*)((char*)d_ws + (size_t)16 * 1024 * 1024); // N_CB floats
    int*   idx = (int*)((char*)d_ws + (size_t)16 * 1024 * 1024 + N_CB * sizeof(float));

    swizzle_cb_kernel<<<(N_CB / 16) * KSTEPS * 32 / 256, 256, 0, stream>>>(cb, swb);
    c2_kernel<<<N_CB / 8, 256, 0, stream>>>(cb, hc2);
    vq_argmin_kernel<<<B_TOK / 64, 128, LDS_BYTES, stream>>>(x, swb, hc2, idx);
    gather_kernel<<<(B_TOK * 128) / 256, 256, 0, stream>>>(emb, idx, out);
}